// _BatchFWPCell_67053029425176
// MI455X (gfx1250) — compile-verified
//
#include <hip/hip_runtime.h>
#include <math.h>

// ---- problem constants (match reference) ----
#define BATCH   2048
#define NQ      12
#define DIM     4096          // 1 << NQ
#define DEPTH   4
#define LAT     512
#define DN      48            // DEPTH * NQ
#define DECAY   0.9f

typedef __attribute__((ext_vector_type(2))) float v2f;
typedef __attribute__((ext_vector_type(8))) float v8f;

// D = A(16x4 f32) * B(4x16 f32) + C(16x16 f32)
static __device__ __forceinline__ v8f wmma_f32x4(v2f a, v2f b, v8f c) {
    return __builtin_amdgcn_wmma_f32_16x16x4_f32(
        /*neg_a=*/false, a, /*neg_b=*/false, b,
        /*c_mod=*/(short)0, c, /*reuse_a=*/false, /*reuse_b=*/false);
}

// Branch-free tanh: tanh(x) = sign(x) * (1 - e^{-2|x|}) / (1 + e^{-2|x|}).
// expf argument always <= 0 -> no overflow, saturates cleanly to +/-1.
// Uses raw v_rcp_f32 (~1 ulp) instead of the IEEE div sequence: single TRANS
// op that co-issues with VALU, no v_div_scale/Newton fixup chain.
static __device__ __forceinline__ float tanh_fast(float x) {
    const float t = __expf(-2.0f * fabsf(x));
    const float r = (1.0f - t) * __builtin_amdgcn_rcpf(1.0f + t);
    return copysignf(r, x);
}

// ---------------------------------------------------------------------------
// Kernel 1: latent = tanh(x @ W_enc^T + b_enc)            [2048 x 512]
//           delta  = latent @ W_upd^T + b_upd             [2048 x 48]
//           fast_next = DECAY * fast_prev + delta  -> d_out fast region
// Block: 256 threads (8 wave32), handles 16 batch rows.
// ---------------------------------------------------------------------------
__global__ __launch_bounds__(256)
void fwp_gemm_kernel(const float* __restrict__ x_t,
                     const float* __restrict__ fast_prev,
                     const float* __restrict__ W_enc,   // [512,12]
                     const float* __restrict__ b_enc,   // [512]
                     const float* __restrict__ W_upd,   // [48,512]
                     const float* __restrict__ b_upd,   // [48]
                     float* __restrict__ fast_out)      // [2048,48]
{
    __shared__ float xt[16 * NQ];          // x tile, row-major
    __shared__ float lat[16 * LAT];        // latent tile (32 KB)
    __shared__ float part[8 * 16 * DN];    // per-wave GEMM2 partials (24 KB)

    const int tid  = threadIdx.x;
    const int wave = tid >> 5;
    const int lane = tid & 31;
    const int m0   = blockIdx.x * 16;

    // load 16x12 x tile
    for (int i = tid; i < 16 * NQ; i += 256) {
        xt[i] = x_t[(m0 + i / NQ) * NQ + (i % NQ)];
    }
    __syncthreads();

    // A-fragment lane mapping (f32 16x4): lanes 0-15 hold K=0,1; 16-31 hold K=2,3
    const int arow  = lane & 15;
    const int khalf = (lane >> 4) << 1;          // 0 or 2
    const int rbase = (lane >> 4) << 3;          // C/D: M = r (+8 for hi lanes)

    // ---- GEMM1 + tanh: 32 N-tiles over 8 waves ----
    for (int t = wave; t < LAT / 16; t += 8) {
        const int col = t * 16 + (lane & 15);    // global latent column
        v8f acc = {0.f, 0.f, 0.f, 0.f, 0.f, 0.f, 0.f, 0.f};
#pragma unroll
        for (int k0 = 0; k0 < NQ; k0 += 4) {
            v2f a, b;
            a.x = xt[arow * NQ + k0 + khalf];
            a.y = xt[arow * NQ + k0 + khalf + 1];
            // B[k][n] = W_enc[n][k]
            b.x = W_enc[col * NQ + k0 + khalf];
            b.y = W_enc[col * NQ + k0 + khalf + 1];
            acc = wmma_f32x4(a, b, acc);
        }
        const float bias = b_enc[col];
#pragma unroll
        for (int r = 0; r < 8; ++r) {
            lat[(r + rbase) * LAT + col] = tanh_fast(acc[r] + bias);
        }
    }
    __syncthreads();

    // ---- GEMM2: each wave covers K-slice [wave*64, wave*64+64) for all 3 tiles ----
    const int kstart = wave * (LAT / 8);
#pragma unroll
    for (int t = 0; t < 3; ++t) {
        const int col = t * 16 + (lane & 15);    // global delta column j
        v8f acc = {0.f, 0.f, 0.f, 0.f, 0.f, 0.f, 0.f, 0.f};
        for (int k0 = kstart; k0 < kstart + LAT / 8; k0 += 4) {
            v2f a, b;
            a.x = lat[arow * LAT + k0 + khalf];
            a.y = lat[arow * LAT + k0 + khalf + 1];
            // B[k][j] = W_upd[j][k]
            b.x = W_upd[col * LAT + k0 + khalf];
            b.y = W_upd[col * LAT + k0 + khalf + 1];
            acc = wmma_f32x4(a, b, acc);
        }
#pragma unroll
        for (int r = 0; r < 8; ++r) {
            part[wave * (16 * DN) + (r + rbase) * DN + col] = acc[r];
        }
    }
    __syncthreads();

    // ---- reduce 8 partials, fuse decay + bias, write fast_next ----
    for (int idx = tid; idx < 16 * DN; idx += 256) {
        const int j = idx % DN;
        float d = b_upd[j];
#pragma unroll
        for (int wv = 0; wv < 8; ++wv) d += part[wv * (16 * DN) + idx];
        const int grow = m0 + idx / DN;
        fast_out[grow * DN + j] = DECAY * fast_prev[grow * DN + j] + d;
    }
}

// ---------------------------------------------------------------------------
// Kernel 2: fast-weight quantum circuit, one block (256 thr) per batch element.
// State (4096 f32 = 16 KB) lives entirely in LDS; 60 RY + 44 CNOT gates;
// Z expectations + readout fused at the end.
// ---------------------------------------------------------------------------
__global__ __launch_bounds__(256)
void fwp_quantum_kernel(const float* __restrict__ x_t,
                        const float* __restrict__ fast_next, // [2048,48] (d_out region)
                        const float* __restrict__ W_ro,      // [1,12]
                        const float* __restrict__ b_ro,      // [1]
                        float* __restrict__ y_out)           // [2048]
{
    __shared__ float st[DIM];
    __shared__ float csC[60], csS[60];
    __shared__ float red[8 * NQ];

    const int b   = blockIdx.x;
    const int tid = threadIdx.x;

    // precompute half-angle cos/sin for all 60 RY gates
    for (int g = tid; g < 60; g += 256) {
        const float th = (g < NQ) ? x_t[b * NQ + g] : fast_next[b * DN + (g - NQ)];
        const float h  = 0.5f * th;
        csC[g] = cosf(h);
        csS[g] = sinf(h);
    }
    // init |+...+> amplitude 2^(-NQ/2)
    for (int i = tid; i < DIM; i += 256) st[i] = 0.015625f;
    __syncthreads();

    auto apply_ry = [&](int w, int g) {
        const float c = csC[g], s = csS[g];
        const int sh     = (NQ - 1) - w;         // pair stride = 1<<sh
        const int stride = 1 << sh;
        for (int p = tid; p < DIM / 2; p += 256) {
            const int rr = p & (stride - 1);
            const int i0 = ((p >> sh) << (sh + 1)) | rr;
            const int i1 = i0 | stride;
            const float a0 = st[i0], a1 = st[i1];
            st[i0] = c * a0 - s * a1;
            st[i1] = s * a0 + c * a1;
        }
        __syncthreads();
    };

    auto apply_cnot = [&](int w) {             // control w, target w+1
        const int sht = (NQ - 2) - w;          // target stride bit
        const int t   = 1 << sht;
        for (int p = tid; p < DIM / 4; p += 256) {
            const int rr = p & (t - 1);
            const int i0 = ((p >> sht) << (sht + 2)) | (1 << (sht + 1)) | rr;
            const int i1 = i0 | t;
            const float a = st[i0], bb = st[i1];
            st[i0] = bb;
            st[i1] = a;
        }
        __syncthreads();
    };

    for (int w = 0; w < NQ; ++w) apply_ry(w, w);
    for (int l = 0; l < DEPTH; ++l) {
        for (int w = 0; w < NQ - 1; ++w) apply_cnot(w);
        for (int w = 0; w < NQ; ++w)     apply_ry(w, NQ + l * NQ + w);
    }

    // Z expectations: q[w] = sum_i st[i]^2 * (bit(11-w,i) ? -1 : +1)
    float q[NQ];
#pragma unroll
    for (int w = 0; w < NQ; ++w) q[w] = 0.f;
    for (int i = tid; i < DIM; i += 256) {
        const float p = st[i] * st[i];
#pragma unroll
        for (int w = 0; w < NQ; ++w)
            q[w] += ((i >> ((NQ - 1) - w)) & 1) ? -p : p;
    }
    // intra-wave reduction (wave32)
#pragma unroll
    for (int w = 0; w < NQ; ++w)
        for (int off = 16; off > 0; off >>= 1)
            q[w] += __shfl_down(q[w], off, 32);

    const int wave = tid >> 5, lane = tid & 31;
    if (lane == 0)
        for (int w = 0; w < NQ; ++w) red[wave * NQ + w] = q[w];
    __syncthreads();

    if (tid == 0) {
        float y = b_ro[0];
        for (int w = 0; w < NQ; ++w) {
            float s = 0.f;
            for (int wv = 0; wv < 8; ++wv) s += red[wv * NQ + w];
            y += s * W_ro[w];
        }
        y_out[b] = y;
    }
}

// ---------------------------------------------------------------------------
extern "C" void kernel_launch(void* const* d_in, const int* in_sizes, int n_in,
                              void* d_out, int out_size, void* d_ws, size_t ws_size,
                              hipStream_t stream) {
    const float* x_t       = (const float*)d_in[0];
    const float* fast_prev = (const float*)d_in[1];
    const float* W_enc     = (const float*)d_in[2];
    const float* b_enc     = (const float*)d_in[3];
    const float* W_upd     = (const float*)d_in[4];
    const float* b_upd     = (const float*)d_in[5];
    const float* W_ro      = (const float*)d_in[6];
    const float* b_ro      = (const float*)d_in[7];

    float* out      = (float*)d_out;
    float* y_t      = out;            // [2048,1]  (tuple element 0)
    float* fast_out = out + BATCH;    // [2048,4,12] (tuple element 1)

    (void)in_sizes; (void)n_in; (void)out_size; (void)d_ws; (void)ws_size;

    fwp_gemm_kernel<<<BATCH / 16, 256, 0, stream>>>(
        x_t, fast_prev, W_enc, b_enc, W_upd, b_upd, fast_out);
    fwp_quantum_kernel<<<BATCH, 256, 0, stream>>>(
        x_t, fast_out, W_ro, b_ro, y_t);
}